// LPModel_85856396248059
// MI455X (gfx1250) — compile-verified
//
#include <hip/hip_runtime.h>
#include <math.h>

// ---------------------------------------------------------------------------
// LPModel on MI455X (gfx1250):
//   per iter: xn = x/(||x||+eps); agg = scatter_add(xn[src] -> dst);
//             x = relu(agg @ W[i])
//   out = sigmoid(x @ final_weight)
//
// Bandwidth-bound: 256 MB edge stream per iteration (HBM), 8 MB xn/agg
// working sets (L2-resident, 192 MB L2). Edge pass: int4 index loads +
// f32 HW atomics. Node transform: V_WMMA_F32_16X16X4_F32 (16 nodes/wave).
// ---------------------------------------------------------------------------

typedef __attribute__((ext_vector_type(2))) float v2f;
typedef __attribute__((ext_vector_type(8))) float v8f;

#define LP_EPS 1e-15f

// fast ~1ulp hardware transcendentals (v_sqrt_f32 / v_rcp_f32 / v_exp_f32)
__device__ __forceinline__ float fast_sqrt(float x) { return __builtin_amdgcn_sqrtf(x); }
__device__ __forceinline__ float fast_rcp(float x)  { return __builtin_amdgcn_rcpf(x); }

// ---------------- normalize: xn = x / (||x|| + eps) ----------------
__global__ void __launch_bounds__(256)
lp_norm_kernel(const float2* __restrict__ x, float2* __restrict__ xn, int n)
{
    int i = blockIdx.x * blockDim.x + threadIdx.x;
    if (i >= n) return;
    float2 v = x[i];
    float inv = fast_rcp(fast_sqrt(v.x * v.x + v.y * v.y) + LP_EPS);
    float2 o;
    o.x = v.x * inv;
    o.y = v.y * inv;
    xn[i] = o;
}

// ---------------- scatter: agg[dst] += xn[src], 4 edges/thread ----------------
__global__ void __launch_bounds__(256)
lp_scatter_kernel(const int4* __restrict__ src4,
                  const int4* __restrict__ dst4,
                  const float2* __restrict__ xn,
                  float* __restrict__ agg,
                  int n_quads)
{
    int i = blockIdx.x * blockDim.x + threadIdx.x;
    if (i >= n_quads) return;

    // Keep the streaming index reads ahead of the atomic pipeline.
    __builtin_prefetch(&src4[i + 4096], 0, 0);   // -> global_prefetch_b8
    __builtin_prefetch(&dst4[i + 4096], 0, 0);

    int4 s = src4[i];   // global_load_b128, fully coalesced
    int4 d = dst4[i];

    float2 v0 = xn[s.x];   // random 8B gathers, L2-resident (8 MB table)
    float2 v1 = xn[s.y];
    float2 v2 = xn[s.z];
    float2 v3 = xn[s.w];

    // HW fp32 atomics (global_atomic_add_f32, no return) into L2-resident agg.
    unsafeAtomicAdd(&agg[2 * d.x + 0], v0.x);
    unsafeAtomicAdd(&agg[2 * d.x + 1], v0.y);
    unsafeAtomicAdd(&agg[2 * d.y + 0], v1.x);
    unsafeAtomicAdd(&agg[2 * d.y + 1], v1.y);
    unsafeAtomicAdd(&agg[2 * d.z + 0], v2.x);
    unsafeAtomicAdd(&agg[2 * d.z + 1], v2.y);
    unsafeAtomicAdd(&agg[2 * d.w + 0], v3.x);
    unsafeAtomicAdd(&agg[2 * d.w + 1], v3.y);
}

// tail for n_edges % 4 != 0 (not hit for 32M edges, kept for generality)
__global__ void lp_scatter_tail_kernel(const int* __restrict__ src,
                                       const int* __restrict__ dst,
                                       const float2* __restrict__ xn,
                                       float* __restrict__ agg,
                                       int start, int n_edges)
{
    int e = start + blockIdx.x * blockDim.x + threadIdx.x;
    if (e >= n_edges) return;
    int s = src[e], d = dst[e];
    float2 v = xn[s];
    unsafeAtomicAdd(&agg[2 * d + 0], v.x);
    unsafeAtomicAdd(&agg[2 * d + 1], v.y);
}

// ---------------- transform: y = relu(agg @ W) via V_WMMA_F32_16X16X4_F32 ----
// MODE 0: epilogue = normalize (produce next iteration's xn)
// MODE 1: epilogue = sigmoid(y . final_weight)  (produce final output)
//
// A (16x4 f32), ISA layout: lane L<16 holds row M=L, VGPR0=K0, VGPR1=K1;
//   lanes 16-31 hold K2,K3 -> set to 0 (W only has 2 rows).
// B (4x16 f32): rows striped across lanes within a VGPR, halves split across
//   lane groups (C/D-analogous): VGPR0 = K0 (lanes 0-15) | K2 (lanes 16-31),
//   VGPR1 = K1 | K3. Only B[0..1][0..1] = W is nonzero.
// D (16x16 f32): VGPR v, lanes 0-15 -> D[v][lane]; lanes 16-31 -> D[v+8][lane-16].
//   => lane 0: y.x of nodes base+0..7, lane 1: y.y; lanes 16/17: nodes base+8..15.
template <int MODE>
__global__ void __launch_bounds__(256)
lp_transform_kernel(const float* __restrict__ agg,
                    const float* __restrict__ W,   // 2x2 row-major (uniform)
                    const float* __restrict__ fw,  // [2], MODE==1 (uniform)
                    float* __restrict__ out,
                    int n_nodes)
{
    const int lane = threadIdx.x & 31;
    const int wave = (blockIdx.x * blockDim.x + threadIdx.x) >> 5;
    const int base = wave * 16;
    if (base >= n_nodes) return;   // wave-uniform exit, EXEC stays all-ones

    // Uniform scalar loads of W (s_load), then per-lane cndmask selection:
    // avoids exec-predicated vector loads for the B operand.
    const float w00 = W[0], w01 = W[1], w10 = W[2], w11 = W[3];

    v2f a;
    const int m = base + lane;
    const bool la = (lane < 16) && (m < n_nodes);
    a.x = la ? agg[2 * m + 0] : 0.0f;   // A[m][0]
    a.y = la ? agg[2 * m + 1] : 0.0f;   // A[m][1]

    v2f b;
    b.x = (lane == 0) ? w00 : (lane == 1) ? w01 : 0.0f;  // B row K=0 (K=2 half: 0)
    b.y = (lane == 0) ? w10 : (lane == 1) ? w11 : 0.0f;  // B row K=1 (K=3 half: 0)

    v8f c = {};
    // 8 args: (neg_a, A, neg_b, B, c_mod, C, reuse_a, reuse_b)
    c = __builtin_amdgcn_wmma_f32_16x16x4_f32(
        false, a, false, b, (short)0, c, false, false);

    float fw0 = 0.0f, fw1 = 0.0f;
    if (MODE == 1) { fw0 = fw[0]; fw1 = fw[1]; }

#pragma unroll
    for (int v = 0; v < 8; ++v) {
        float yv = fmaxf(c[v], 0.0f);        // relu, own component
        float yo = __shfl_xor(yv, 1, 32);    // partner component (lane0<->1, 16<->17)
        int node = base + ((lane >> 4) << 3) + v;
        bool active = ((lane & 14) == 0) && (node < n_nodes); // lanes 0,1,16,17
        if (MODE == 0) {
            float inv = fast_rcp(fast_sqrt(yv * yv + yo * yo) + LP_EPS);
            if (active) out[2 * node + (lane & 1)] = yv * inv;
        } else {
            float t = yv * fw0 + yo * fw1;   // valid on even lanes (yv = y.x)
            float sg = fast_rcp(1.0f + __expf(-t));
            if (active && ((lane & 1) == 0)) out[node] = sg;
        }
    }
}

// ---------------------------------------------------------------------------
extern "C" void kernel_launch(void* const* d_in, const int* in_sizes, int n_in,
                              void* d_out, int out_size, void* d_ws, size_t ws_size,
                              hipStream_t stream)
{
    const float* x  = (const float*)d_in[0];   // [n_nodes, 2]
    const int*   ei = (const int*)d_in[1];     // [2, n_edges]
    const float* W  = (const float*)d_in[2];   // [n_iter, 2, 2]
    const float* fw = (const float*)d_in[3];   // [2]

    const int n_nodes = in_sizes[0] / 2;
    const int n_edges = in_sizes[1] / 2;
    const int n_iter  = in_sizes[2] / 4;

    const int* src = ei;
    const int* dst = ei + n_edges;

    float* xn  = (float*)d_ws;                       // 2*n_nodes floats
    float* agg = xn + (size_t)2 * n_nodes;           // 2*n_nodes floats

    const int n_quads = n_edges >> 2;
    const int n_tail  = n_edges & 3;

    const dim3 blk(256);
    const dim3 g_nodes((n_nodes + 255) / 256);
    const dim3 g_quads((n_quads + 255) / 256);
    const dim3 g_xform((n_nodes + 127) / 128);       // 8 waves * 16 nodes / block

    lp_norm_kernel<<<g_nodes, blk, 0, stream>>>((const float2*)x, (float2*)xn, n_nodes);

    for (int it = 0; it < n_iter; ++it) {
        hipMemsetAsync(agg, 0, (size_t)2 * n_nodes * sizeof(float), stream);
        lp_scatter_kernel<<<g_quads, blk, 0, stream>>>(
            (const int4*)src, (const int4*)dst, (const float2*)xn, agg, n_quads);
        if (n_tail)
            lp_scatter_tail_kernel<<<1, 64, 0, stream>>>(
                src, dst, (const float2*)xn, agg, n_quads * 4, n_edges);

        if (it + 1 < n_iter) {
            // fused relu + re-normalize into xn for the next iteration
            lp_transform_kernel<0><<<g_xform, blk, 0, stream>>>(
                agg, W + 4 * it, nullptr, xn, n_nodes);
        } else {
            // fused relu + sigmoid(x . final_weight) into d_out
            lp_transform_kernel<1><<<g_xform, blk, 0, stream>>>(
                agg, W + 4 * it, fw, (float*)d_out, n_nodes);
        }
    }
}